// LocalMPNN_88493506166790
// MI455X (gfx1250) — compile-verified
//
#include <hip/hip_runtime.h>
#include <hip/hip_bf16.h>

typedef float v2f __attribute__((ext_vector_type(2)));
typedef float v8f __attribute__((ext_vector_type(8)));

#define DIM   128
#define WPAD  132   // padded LDS row stride (floats): bank = (4*row + k) % 64 -> conflict-free
#define XPAD  132

// ---------------------------------------------------------------------------
// out[row, col] = act( sum_f A[row, f] * W[col, f] + bias[col] )
// A: N x 128, W: 128 x 128 (row-major, W[g][f]), out: N x 128
// One block: 16 rows. 8 waves; wave w computes the 16-column tile w.
// K-loop of 32 x V_WMMA_F32_16X16X4_F32 (native fp32 matrix path).
// ---------------------------------------------------------------------------
__global__ __launch_bounds__(256)
void gemm_bias_wmma(const float* __restrict__ A,
                    const float* __restrict__ W,
                    const float* __restrict__ bias,
                    float* __restrict__ out,
                    int N, int do_relu)
{
    extern __shared__ float smem[];
    float* sW = smem;                 // 128 * WPAD floats
    float* sX = smem + DIM * WPAD;    // 16  * XPAD floats

    const int t    = threadIdx.x;
    const int row0 = blockIdx.x * 16;

    // Stage W (128x128) into LDS, float4-coalesced, padded rows.
    for (int i = t; i < DIM * 32; i += 256) {
        const int r  = i >> 5;
        const int c4 = (i & 31) << 2;
        const float4 v = *(const float4*)(W + r * DIM + c4);
        float* p = sW + r * WPAD + c4;
        p[0] = v.x; p[1] = v.y; p[2] = v.z; p[3] = v.w;
    }
    // Stage the 16x128 A tile into LDS.
    for (int i = t; i < 16 * 32; i += 256) {
        const int r  = i >> 5;
        const int c4 = (i & 31) << 2;
        int rr = row0 + r; if (rr >= N) rr = N - 1;   // clamp (N is a multiple of 16 here)
        const float4 v = *(const float4*)(A + (long)rr * DIM + c4);
        float* p = sX + r * XPAD + c4;
        p[0] = v.x; p[1] = v.y; p[2] = v.z; p[3] = v.w;
    }
    __syncthreads();

    const int wave    = t >> 5;          // 0..7 -> column tile
    const int lane    = t & 31;
    const int l16     = lane & 15;
    const int hi      = lane >> 4;       // 0: K pair {0,1}; 1: K pair {2,3}
    const int colBase = wave * 16;

    // Seed accumulator with the bias (each output element gets it exactly once).
    const float b = bias[colBase + l16];
    v8f c;
#pragma unroll
    for (int v = 0; v < 8; ++v) c[v] = b;

    // D = A x B + C over K=128 in steps of 4.
    // A frag (16x4): lane<16 -> {A[l16][k0], A[l16][k0+1]}, lane>=16 -> K+2.
    // B frag (4x16): B[k][g] = W[g][k]; mirrored lane layout.
#pragma unroll
    for (int k0 = 0; k0 < DIM; k0 += 4) {
        v2f a, bf;
        const int ka = k0 + 2 * hi;
        a.x  = sX[l16 * XPAD + ka + 0];
        a.y  = sX[l16 * XPAD + ka + 1];
        bf.x = sW[(colBase + l16) * WPAD + ka + 0];
        bf.y = sW[(colBase + l16) * WPAD + ka + 1];
        c = __builtin_amdgcn_wmma_f32_16x16x4_f32(
                /*neg_a=*/false, a, /*neg_b=*/false, bf,
                /*c_mod=*/(short)0, c, /*reuse_a=*/false, /*reuse_b=*/false);
    }

    // C/D layout: lane gives col (l16), VGPR v gives row v + 8*hi.
#pragma unroll
    for (int v = 0; v < 8; ++v) {
        const int r  = v + 8 * hi;
        const int rr = row0 + r;
        float val = c[v];
        if (do_relu) val = fmaxf(val, 0.0f);
        if (rr < N) out[(long)rr * DIM + colBase + l16] = val;
    }
}

// agg = (1 + eps) * X   (vectorized float4; eps read on-device, graph-safe)
__global__ __launch_bounds__(256)
void scale_copy(const float* __restrict__ X, const float* __restrict__ eps,
                float* __restrict__ agg, long n4)
{
    const long i = (long)blockIdx.x * blockDim.x + threadIdx.x;
    if (i >= n4) return;
    const float s = 1.0f + eps[0];
    float4 v = ((const float4*)X)[i];
    v.x *= s; v.y *= s; v.z *= s; v.w *= s;
    ((float4*)agg)[i] = v;
}

// For each edge e: agg[src[e], :] += msg[dst[e], :]
// 32 lanes per edge, float4 gather + 4 scoped f32 atomic adds per lane.
__global__ __launch_bounds__(256)
void edge_scatter(const float* __restrict__ msg,
                  const long long* __restrict__ edge_index,
                  float* __restrict__ agg, int E)
{
    const long tid  = (long)blockIdx.x * blockDim.x + threadIdx.x;
    const long edge = tid >> 5;
    const int  lane = (int)(tid & 31);
    if (edge >= E) return;

    const long long s = edge_index[edge];       // src row (first E entries)
    const long long d = edge_index[(long)E + edge]; // dst row (second E entries)

    const float4 v = *(const float4*)(msg + (long)d * DIM + lane * 4);
    float* a = agg + (long)s * DIM + lane * 4;
    __hip_atomic_fetch_add(a + 0, v.x, __ATOMIC_RELAXED, __HIP_MEMORY_SCOPE_AGENT);
    __hip_atomic_fetch_add(a + 1, v.y, __ATOMIC_RELAXED, __HIP_MEMORY_SCOPE_AGENT);
    __hip_atomic_fetch_add(a + 2, v.z, __ATOMIC_RELAXED, __HIP_MEMORY_SCOPE_AGENT);
    __hip_atomic_fetch_add(a + 3, v.w, __ATOMIC_RELAXED, __HIP_MEMORY_SCOPE_AGENT);
}

extern "C" void kernel_launch(void* const* d_in, const int* in_sizes, int n_in,
                              void* d_out, int out_size, void* d_ws, size_t ws_size,
                              hipStream_t stream)
{
    const float*     X     = (const float*)d_in[0];
    const long long* ei    = (const long long*)d_in[1];  // int64 per reference
    const float*     eps   = (const float*)d_in[2];
    const float*     msg_w = (const float*)d_in[3];
    const float*     msg_b = (const float*)d_in[4];
    const float*     lin_w = (const float*)d_in[5];
    const float*     lin_b = (const float*)d_in[6];

    const int N = in_sizes[0] / DIM;   // 50000
    const int E = in_sizes[1] / 2;     // 625000

    float* msg = (float*)d_ws;                    // N * 128 floats
    float* agg = msg + (size_t)N * DIM;           // N * 128 floats
    float* out = (float*)d_out;

    const int    nblk = (N + 15) / 16;
    const size_t lds  = (size_t)(DIM * WPAD + 16 * XPAD) * sizeof(float); // ~74 KB

    // 1) msg = X @ msg_w^T + msg_b
    gemm_bias_wmma<<<nblk, 256, lds, stream>>>(X, msg_w, msg_b, msg, N, 0);

    // 2) agg = (1 + eps) * X
    const long n4 = (long)N * DIM / 4;
    scale_copy<<<(int)((n4 + 255) / 256), 256, 0, stream>>>(X, eps, agg, n4);

    // 3) agg[src] += msg[dst]  (atomic scatter-add, the bandwidth-dominant phase)
    const long threads = (long)E * 32;
    edge_scatter<<<(int)((threads + 255) / 256), 256, 0, stream>>>(msg, ei, agg, E);

    // 4) out = relu(agg @ lin_w^T + lin_b)
    gemm_bias_wmma<<<nblk, 256, lds, stream>>>(agg, lin_w, lin_b, out, N, 1);
}